// SparseLinear_76751065579575
// MI455X (gfx1250) — compile-verified
//
#include <hip/hip_runtime.h>
#include <hip/hip_bf16.h>
#include <cstdint>

// SpMM via WMMA: out[8192,512] = sparse(A) @ W + bias
// f32 16x16x4 WMMA; selector-matrix formulation of the segmented scatter-add.

typedef float v2f __attribute__((ext_vector_type(2)));
typedef float v8f __attribute__((ext_vector_type(8)));

#define M_ROWS   8192
#define OUT_F    512
#define TILE_M   16
#define CHUNK    512      // nnz staged in LDS per iteration (4 KB LDS)
#define THREADS  256      // 8 waves (wave32)

__global__ __launch_bounds__(THREADS)
void spmm_wmma_f32_kernel(const int* __restrict__ rows,     // [nnz] sorted
                          const int* __restrict__ cols,     // [nnz]
                          const float* __restrict__ vals,   // [nnz]
                          const float* __restrict__ weight, // [N, OUT] row-major
                          const float* __restrict__ bias,   // [OUT]
                          float* __restrict__ out,          // [M, OUT]
                          int nnz)
{
    // packed meta: bits[26:0] = col, bits[31:27] = local row (16 == padding sentinel)
    __shared__ uint32_t s_meta[CHUNK];
    __shared__ float    s_val[CHUNK];

    const int tile  = blockIdx.x;          // 0..511
    const int row0  = tile * TILE_M;
    const int tid   = threadIdx.x;
    const int lane  = tid & 31;
    const int wave  = tid >> 5;            // 0..7
    const int m     = lane & 15;           // M-row (A) / N-col (B,C) within 16
    const int half  = lane >> 4;           // 0: K{0,1}   1: K{2,3}
    const int nbase = wave * 64;           // this wave's 64 output columns

    // --- binary search segment bounds in sorted rows[] (uniform per block) ---
    int lo = 0, hi = nnz;
    while (lo < hi) { int mid = (lo + hi) >> 1; if (rows[mid] < row0)          lo = mid + 1; else hi = mid; }
    const int seg_lo = lo;
    hi = nnz;
    while (lo < hi) { int mid = (lo + hi) >> 1; if (rows[mid] < row0 + TILE_M) lo = mid + 1; else hi = mid; }
    const int seg_hi = lo;

    const v8f zero = {0.f,0.f,0.f,0.f,0.f,0.f,0.f,0.f};
    v8f acc[4];
#pragma unroll
    for (int t = 0; t < 4; ++t) acc[t] = zero;

    for (int base = seg_lo; base < seg_hi; base += CHUNK) {
        const int cnt  = (seg_hi - base < CHUNK) ? (seg_hi - base) : CHUNK;
        const int cnt4 = (cnt + 3) & ~3;

        __syncthreads();   // previous chunk fully consumed
        for (int i = tid; i < cnt; i += THREADS) {
            const int g = base + i;
            const uint32_t rl = (uint32_t)(rows[g] - row0);      // 0..15
            s_meta[i] = ((uint32_t)cols[g]) | (rl << 27);
            s_val[i]  = vals[g];
        }
        // pad the ragged tail with never-matching entries (rl=16, col=0, val=0)
        for (int i = cnt + tid; i < cnt4; i += THREADS) {
            s_meta[i] = (16u << 27);
            s_val[i]  = 0.f;
        }
        __syncthreads();

        // uniform k-loop: EXEC stays all-ones around the WMMAs
        for (int k = 0; k < cnt4; k += 4) {
            const int e0 = k + 2 * half;   // this half-wave's K pair
            const int e1 = e0 + 1;

            const uint32_t p0 = s_meta[e0];
            const uint32_t p1 = s_meta[e1];
            const float    v0 = s_val[e0];
            const float    v1 = s_val[e1];
            const int col0 = (int)(p0 & 0x07FFFFFFu);
            const int col1 = (int)(p1 & 0x07FFFFFFu);
            const int rl0  = (int)(p0 >> 27);
            const int rl1  = (int)(p1 >> 27);

            // A fragment (16x4 f32): VGPR0 = K{0|2}, VGPR1 = K{1|3}
            v2f a;
            a.x = (rl0 == m) ? v0 : 0.f;
            a.y = (rl1 == m) ? v1 : 0.f;

            // B fragments (4x16 f32), one per 16-col N-tile; 64B-contiguous per half-wave
            const float* w0 = weight + (size_t)col0 * OUT_F + nbase + m;
            const float* w1 = weight + (size_t)col1 * OUT_F + nbase + m;

            v2f b;
            b.x = w0[0];  b.y = w1[0];
            acc[0] = __builtin_amdgcn_wmma_f32_16x16x4_f32(false, a, false, b,
                                                           (short)0, acc[0], false, false);
            b.x = w0[16]; b.y = w1[16];
            acc[1] = __builtin_amdgcn_wmma_f32_16x16x4_f32(false, a, false, b,
                                                           (short)0, acc[1], false, false);
            b.x = w0[32]; b.y = w1[32];
            acc[2] = __builtin_amdgcn_wmma_f32_16x16x4_f32(false, a, false, b,
                                                           (short)0, acc[2], false, false);
            b.x = w0[48]; b.y = w1[48];
            acc[3] = __builtin_amdgcn_wmma_f32_16x16x4_f32(false, a, false, b,
                                                           (short)0, acc[3], false, false);
        }
    }

    // --- store D + bias; C/D layout: VGPR v -> row0+v (lanes 0-15) / row0+v+8 (lanes 16-31)
    // Non-temporal stores: keep the 128 MB weight matrix resident in the 192 MB L2.
#pragma unroll
    for (int t = 0; t < 4; ++t) {
        const int coln = nbase + t * 16 + m;
        const float bv = bias[coln];
#pragma unroll
        for (int v = 0; v < 8; ++v) {
            const int r = row0 + v + half * 8;
            __builtin_nontemporal_store(acc[t][v] + bv, &out[(size_t)r * OUT_F + coln]);
        }
    }
}

extern "C" void kernel_launch(void* const* d_in, const int* in_sizes, int n_in,
                              void* d_out, int out_size, void* d_ws, size_t ws_size,
                              hipStream_t stream) {
    // setup_inputs order: indices[2,NNZ] (int32), values[NNZ] (f32), m, n,
    //                     weight[N,OUT] (f32), bias[OUT] (f32)
    const int*   indices = (const int*)d_in[0];
    const float* values  = (const float*)d_in[1];
    const float* weight  = (const float*)d_in[4];
    const float* bias    = (const float*)d_in[5];
    float*       out     = (float*)d_out;

    const int nnz  = in_sizes[1];
    const int* rows = indices;        // indices[0, :]
    const int* cols = indices + nnz;  // indices[1, :]

    dim3 grid(M_ROWS / TILE_M);       // 512 tiles of 16 rows
    dim3 block(THREADS);
    spmm_wmma_f32_kernel<<<grid, block, 0, stream>>>(rows, cols, values,
                                                     weight, bias, out, nnz);
}